// Net_16801912062043
// MI455X (gfx1250) — compile-verified
//
#include <hip/hip_runtime.h>
#include <cmath>

#define NN   100000
#define EE   3200000
#define F_INd 500
#define HID  16
#define NC   3

typedef __attribute__((ext_vector_type(2))) float v2f;
typedef __attribute__((ext_vector_type(8))) float v8f;

// ---------------- degree / normalization ----------------
__global__ void k_init_deg(float* __restrict__ deg) {
  int i = blockIdx.x * blockDim.x + threadIdx.x;
  if (i < NN) deg[i] = 1.0f;              // self-loop
}

__global__ void k_count_deg(const int* __restrict__ dst, float* __restrict__ deg) {
  int e = blockIdx.x * blockDim.x + threadIdx.x;
  if (e < EE) atomicAdd(&deg[dst[e]], 1.0f);
}

__global__ void k_rsqrt(float* __restrict__ dinv) {
  int i = blockIdx.x * blockDim.x + threadIdx.x;
  if (i < NN) dinv[i] = rsqrtf(dinv[i]);  // deg >= 1 always
}

// ---------------- GEMM1: X[N,500] @ W1[500,16] via FP32 WMMA ----------------
__global__ __launch_bounds__(256) void k_gemm1_wmma(const float* __restrict__ X,
                                                    const float* __restrict__ W,
                                                    float* __restrict__ H1) {
  __shared__ float Wlds[F_INd * HID];     // 32 KB of the 320 KB WGP LDS
  for (int i = threadIdx.x; i < F_INd * HID; i += 256) Wlds[i] = W[i];
  __syncthreads();

  const int wave = threadIdx.x >> 5;
  const int lane = threadIdx.x & 31;
  const int tile = blockIdx.x * 8 + wave;          // 16 output rows per wave
  if (tile >= NN / 16) return;                     // whole-wave exit: EXEC stays all-1s

  const int row_base = tile * 16;
  const int m    = lane & 15;                      // A: row M / B,D: col N
  const int hi   = lane >> 4;                      // lane half
  const int koff = hi << 1;                        // K sub-offset {0,2}

  const float* xrow = X + (size_t)(row_base + m) * F_INd;

  v8f acc = {};
  for (int k0 = 0; k0 < F_INd; k0 += 4) {          // 125 WMMA ops, K exact
    v2f a, b;
    a.x = xrow[k0 + koff];
    a.y = xrow[k0 + koff + 1];
    b.x = Wlds[(k0 + koff)     * HID + m];
    b.y = Wlds[(k0 + koff + 1) * HID + m];
    acc = __builtin_amdgcn_wmma_f32_16x16x4_f32(false, a, false, b,
                                                (short)0, acc, false, false);
  }

  const int rbase = row_base + (hi << 3);          // D: M = r + 8*hi
#pragma unroll
  for (int r = 0; r < 8; ++r)
    H1[(size_t)(rbase + r) * HID + m] = acc[r];
}

// ---------------- layer-1 aggregation ----------------
__global__ void k_agg1_init(const float* __restrict__ h1, const float* __restrict__ dinv,
                            float* __restrict__ agg1) {
  int i = blockIdx.x * blockDim.x + threadIdx.x;   // over N*16
  if (i >= NN * HID) return;
  int node = i >> 4;
  float di = dinv[node];
  agg1[i] = h1[i] * di * di;                       // self-loop term
}

__global__ void k_agg1_edges(const float* __restrict__ h1, const float* __restrict__ dinv,
                             const int* __restrict__ src, const int* __restrict__ dst,
                             float* __restrict__ agg1) {
  long long idx = (long long)blockIdx.x * blockDim.x + threadIdx.x;  // E*16 threads
  int e = (int)(idx >> 4);
  int f = (int)(idx & 15);
  if (e >= EE) return;
  int s = src[e], d = dst[e];
  float nrm = dinv[s] * dinv[d];
  atomicAdd(&agg1[(size_t)d * HID + f], h1[(size_t)s * HID + f] * nrm);
}

// ---------------- bias + ReLU + tiny GEMM2 (16x3) ----------------
__global__ void k_relu_lin2(const float* __restrict__ agg1, const float* __restrict__ b1,
                            const float* __restrict__ W2, float* __restrict__ h2) {
  int i = blockIdx.x * blockDim.x + threadIdx.x;
  if (i >= NN) return;
  float v[HID];
#pragma unroll
  for (int f = 0; f < HID; ++f) {
    float t = agg1[(size_t)i * HID + f] + b1[f];
    v[f] = t > 0.0f ? t : 0.0f;
  }
#pragma unroll
  for (int c = 0; c < NC; ++c) {
    float s = 0.0f;
#pragma unroll
    for (int f = 0; f < HID; ++f) s += v[f] * W2[f * NC + c];
    h2[(size_t)i * 4 + c] = s;
  }
  h2[(size_t)i * 4 + 3] = 0.0f;
}

// ---------------- layer-2 aggregation ----------------
__global__ void k_agg2_init(const float* __restrict__ h2, const float* __restrict__ dinv,
                            float* __restrict__ agg2) {
  int i = blockIdx.x * blockDim.x + threadIdx.x;   // over N*4
  if (i >= NN * 4) return;
  int node = i >> 2;
  float di = dinv[node];
  agg2[i] = h2[i] * di * di;
}

__global__ void k_agg2_edges(const float* __restrict__ h2, const float* __restrict__ dinv,
                             const int* __restrict__ src, const int* __restrict__ dst,
                             float* __restrict__ agg2) {
  int e = blockIdx.x * blockDim.x + threadIdx.x;
  if (e >= EE) return;
  int s = src[e], d = dst[e];
  float nrm = dinv[s] * dinv[d];
#pragma unroll
  for (int c = 0; c < NC; ++c)
    atomicAdd(&agg2[(size_t)d * 4 + c], h2[(size_t)s * 4 + c] * nrm);
}

// ---------------- bias + log_softmax ----------------
__global__ void k_logsoftmax(const float* __restrict__ agg2, const float* __restrict__ b2,
                             float* __restrict__ out) {
  int i = blockIdx.x * blockDim.x + threadIdx.x;
  if (i >= NN) return;
  float z0 = agg2[(size_t)i * 4 + 0] + b2[0];
  float z1 = agg2[(size_t)i * 4 + 1] + b2[1];
  float z2 = agg2[(size_t)i * 4 + 2] + b2[2];
  float mx = fmaxf(z0, fmaxf(z1, z2));
  float lse = logf(expf(z0 - mx) + expf(z1 - mx) + expf(z2 - mx));
  out[(size_t)i * NC + 0] = z0 - mx - lse;
  out[(size_t)i * NC + 1] = z1 - mx - lse;
  out[(size_t)i * NC + 2] = z2 - mx - lse;
}

extern "C" void kernel_launch(void* const* d_in, const int* in_sizes, int n_in,
                              void* d_out, int out_size, void* d_ws, size_t ws_size,
                              hipStream_t stream) {
  const float* x  = (const float*)d_in[0];
  const float* W1 = (const float*)d_in[1];
  const float* b1 = (const float*)d_in[2];
  const float* W2 = (const float*)d_in[3];
  const float* b2 = (const float*)d_in[4];
  const int*   ei = (const int*)d_in[5];
  const int* src = ei;
  const int* dst = ei + EE;
  float* out = (float*)d_out;

  // workspace layout (all rewritten every call; 41*N floats = 16.4 MB)
  float* ws   = (float*)d_ws;
  float* h1   = ws;                          // N*16
  float* dinv = h1   + (size_t)NN * HID;     // N   (deg -> dinv in place)
  float* agg1 = dinv + (size_t)NN;           // N*16
  float* h2   = agg1 + (size_t)NN * HID;     // N*4
  float* agg2 = h2   + (size_t)NN * 4;       // N*4

  const int T = 256;
  k_init_deg   <<<(NN + T - 1) / T, T, 0, stream>>>(dinv);
  k_count_deg  <<<(EE + T - 1) / T, T, 0, stream>>>(dst, dinv);
  k_rsqrt      <<<(NN + T - 1) / T, T, 0, stream>>>(dinv);

  k_gemm1_wmma <<<(NN / 16 + 7) / 8, T, 0, stream>>>(x, W1, h1);

  k_agg1_init  <<<(NN * HID + T - 1) / T, T, 0, stream>>>(h1, dinv, agg1);
  {
    long long total = (long long)EE * HID;
    int blocks = (int)((total + T - 1) / T);
    k_agg1_edges<<<blocks, T, 0, stream>>>(h1, dinv, src, dst, agg1);
  }

  k_relu_lin2  <<<(NN + T - 1) / T, T, 0, stream>>>(agg1, b1, W2, h2);

  k_agg2_init  <<<(NN * 4 + T - 1) / T, T, 0, stream>>>(h2, dinv, agg2);
  k_agg2_edges <<<(EE + T - 1) / T, T, 0, stream>>>(h2, dinv, src, dst, agg2);

  k_logsoftmax <<<(NN + T - 1) / T, T, 0, stream>>>(agg2, b2, out);
}